// GNNMultiEdgeset_13589276524783
// MI455X (gfx1250) — compile-verified
//
#include <hip/hip_runtime.h>
#include <cstddef>

// ---------------------------------------------------------------------------
// Problem constants (from reference setup_inputs)
// ---------------------------------------------------------------------------
#define NN   20000   // nodes (divisible by 16)
#define EE   60000   // edges (divisible by 16)
#define GG   64      // graphs
#define RC   8       // R*C replicas
#define FIN  64      // input node features
#define FE   16      // edge features
#define HH   128     // hidden
#define OO   32      // output features
#define NLAY 2

typedef __attribute__((ext_vector_type(2))) float v2f;
typedef __attribute__((ext_vector_type(8))) float v8f;

__device__ __forceinline__ float gelu_f(float x) {
  // exact GELU: 0.5*x*(1+erf(x/sqrt(2)))
  return 0.5f * x * (1.0f + erff(x * 0.70710678118654752f));
}

// ---------------------------------------------------------------------------
// One 16x16 fp32 WMMA tile accumulation: C += A(16xK) * B(Kx16)
// A row-major with leading dim lda, B row-major with leading dim ldb.
// Fragment layouts per CDNA5 ISA 7.12.2:
//   A 16x4: lane L holds A[L%16][2*(L/16) + {0,1}]
//   B 4x16: lane L holds B[2*(L/16) + {0,1}][L%16]
//   C/D:    c[v] is element (row = v + 8*(L/16), col = L%16)
// ---------------------------------------------------------------------------
template <int K>
__device__ __forceinline__ v8f wmma_acc(const float* A, int lda,
                                        const float* B, int ldb, v8f c) {
  const int lane = threadIdx.x & 31;
  const int mn = lane & 15;          // A row / B col for this lane
  const int kh = (lane >> 4) << 1;   // 0 for lanes 0-15, 2 for lanes 16-31
#pragma unroll
  for (int k = 0; k < K; k += 4) {
    v2f a, b;
    a[0] = A[mn * lda + k + kh + 0];
    a[1] = A[mn * lda + k + kh + 1];
    b[0] = B[(k + kh + 0) * ldb + mn];
    b[1] = B[(k + kh + 1) * ldb + mn];
    c = __builtin_amdgcn_wmma_f32_16x16x4_f32(false, a, false, b, (short)0, c,
                                              false, false);
  }
  return c;
}

// ---------------------------------------------------------------------------
// Kernel 1: atom encoder  x = x_feat @ atom_W + atom_b, broadcast to RC slices
// grid = NN/16 blocks, 256 threads (8 waves -> 8 column tiles of H=128)
// ---------------------------------------------------------------------------
__global__ __launch_bounds__(256) void atom_gemm_kernel(
    const float* __restrict__ xf, const float* __restrict__ W,
    const float* __restrict__ bias, float* __restrict__ ux) {
  const int row0 = blockIdx.x * 16;
  const int wave = threadIdx.x >> 5;
  const int col0 = wave * 16;
  v8f c = {};
  c = wmma_acc<FIN>(xf + (size_t)row0 * FIN, FIN, W + col0, HH, c);
  const int lane = threadIdx.x & 31;
  const int n = lane & 15;
  const int kh = lane >> 4;
  const float bv = bias[col0 + n];
#pragma unroll
  for (int v = 0; v < 8; ++v) {
    const float val = c[v] + bv;
    const size_t row = (size_t)(row0 + v + 8 * kh);
#pragma unroll
    for (int rc = 0; rc < RC; ++rc)
      ux[(size_t)rc * NN * HH + row * HH + col0 + n] = val;
  }
}

// ---------------------------------------------------------------------------
// Kernel 2: edge embedding  e = edge_attr @ bond_W[l] + bond_b[l]
// grid = EE/16 blocks, 256 threads
// ---------------------------------------------------------------------------
__global__ __launch_bounds__(256) void edge_gemm_kernel(
    const float* __restrict__ ea, const float* __restrict__ W,
    const float* __restrict__ bias, float* __restrict__ eemb) {
  const int row0 = blockIdx.x * 16;
  const int wave = threadIdx.x >> 5;
  const int col0 = wave * 16;
  v8f c = {};
  c = wmma_acc<FE>(ea + (size_t)row0 * FE, FE, W + col0, HH, c);
  const int lane = threadIdx.x & 31;
  const int n = lane & 15;
  const int kh = lane >> 4;
  const float bv = bias[col0 + n];
#pragma unroll
  for (int v = 0; v < 8; ++v)
    eemb[(size_t)(row0 + v + 8 * kh) * HH + col0 + n] = c[v] + bv;
}

// ---------------------------------------------------------------------------
// Kernel 3: message + scatter-add
//   msg = gelu(ux[rc,src] + eemb[e]) * mask[rc,src]*mask[rc,dst]
//   agg[rc,dst] += msg      (fp32 HW atomics)
// Each group of 32 threads covers one (rc, edge): 32 chunks of float4.
// ---------------------------------------------------------------------------
__global__ __launch_bounds__(256) void msg_scatter_kernel(
    const float* __restrict__ ux, const float* __restrict__ eemb,
    const int* __restrict__ src, const int* __restrict__ dst,
    const float* __restrict__ mask, float* __restrict__ agg) {
  const long long tid = (long long)blockIdx.x * 256 + threadIdx.x;
  if (tid >= (long long)RC * EE * (HH / 4)) return;
  const int chunk = (int)(tid & 31);
  const int e = (int)((tid >> 5) % EE);
  const int rc = (int)((tid >> 5) / EE);
  const int s = src[e];
  const int d = dst[e];
  const float mk = mask[(size_t)rc * NN + s] * mask[(size_t)rc * NN + d];
  const float4 xs =
      *(const float4*)(ux + (size_t)rc * NN * HH + (size_t)s * HH + chunk * 4);
  const float4 ee = *(const float4*)(eemb + (size_t)e * HH + chunk * 4);
  float* ap = agg + (size_t)rc * NN * HH + (size_t)d * HH + chunk * 4;
  unsafeAtomicAdd(ap + 0, gelu_f(xs.x + ee.x) * mk);
  unsafeAtomicAdd(ap + 1, gelu_f(xs.y + ee.y) * mk);
  unsafeAtomicAdd(ap + 2, gelu_f(xs.z + ee.z) * mk);
  unsafeAtomicAdd(ap + 3, gelu_f(xs.w + ee.w) * mk);
}

// ---------------------------------------------------------------------------
// Kernel 4: fused GIN node update (in place on ux)
//   h  = (1+eps)*ux + agg
//   ux = gelu( gelu(h @ W1 + b1) @ W2 + b2 )
// grid = RC * NN/16 blocks, 256 threads (8 waves = 8 column tiles)
// ---------------------------------------------------------------------------
__global__ __launch_bounds__(256) void gin_mlp_kernel(
    float* __restrict__ ux, const float* __restrict__ agg,
    const float* __restrict__ eps_l, const float* __restrict__ W1,
    const float* __restrict__ b1, const float* __restrict__ W2,
    const float* __restrict__ b2) {
  __shared__ float hA[16 * 132];
  __shared__ float hB[16 * 132];
  const int row_tile = blockIdx.x % (NN / 16);
  const int rc = blockIdx.x / (NN / 16);
  const int row0 = row_tile * 16;
  const float one_eps = 1.0f + eps_l[0];
  float* uxp = ux + (size_t)rc * NN * HH;
  const float* agp = agg + (size_t)rc * NN * HH;

  for (int i = threadIdx.x; i < 16 * HH; i += 256) {
    const int r = i >> 7;
    const int cc = i & 127;
    const size_t g = (size_t)(row0 + r) * HH + cc;
    hA[r * 132 + cc] = one_eps * uxp[g] + agp[g];
  }
  __syncthreads();

  const int wave = threadIdx.x >> 5;
  const int col0 = wave * 16;
  const int lane = threadIdx.x & 31;
  const int n = lane & 15;
  const int kh = lane >> 4;

  v8f c = {};
  c = wmma_acc<HH>(hA, 132, W1 + col0, HH, c);
  {
    const float bv = b1[col0 + n];
#pragma unroll
    for (int v = 0; v < 8; ++v)
      hB[(v + 8 * kh) * 132 + col0 + n] = gelu_f(c[v] + bv);
  }
  __syncthreads();

  v8f c2 = {};
  c2 = wmma_acc<HH>(hB, 132, W2 + col0, HH, c2);
  {
    const float bv = b2[col0 + n];
#pragma unroll
    for (int v = 0; v < 8; ++v)
      uxp[(size_t)(row0 + v + 8 * kh) * HH + col0 + n] = gelu_f(c2[v] + bv);
  }
}

// ---------------------------------------------------------------------------
// Kernel 5: head  out = gelu(ux @ h1W + h1b) @ h2W + h2b   -> obuf [RC,N,O]
// ---------------------------------------------------------------------------
__global__ __launch_bounds__(256) void head_mlp_kernel(
    const float* __restrict__ ux, const float* __restrict__ W1,
    const float* __restrict__ b1, const float* __restrict__ W2,
    const float* __restrict__ b2, float* __restrict__ obuf) {
  __shared__ float hA[16 * 132];
  __shared__ float hB[16 * 132];
  const int row_tile = blockIdx.x % (NN / 16);
  const int rc = blockIdx.x / (NN / 16);
  const int row0 = row_tile * 16;
  const float* uxp = ux + (size_t)rc * NN * HH;

  for (int i = threadIdx.x; i < 16 * HH; i += 256) {
    const int r = i >> 7;
    const int cc = i & 127;
    hA[r * 132 + cc] = uxp[(size_t)(row0 + r) * HH + cc];
  }
  __syncthreads();

  const int wave = threadIdx.x >> 5;
  const int col0 = wave * 16;
  const int lane = threadIdx.x & 31;
  const int n = lane & 15;
  const int kh = lane >> 4;

  v8f c = {};
  c = wmma_acc<HH>(hA, 132, W1 + col0, HH, c);
  {
    const float bv = b1[col0 + n];
#pragma unroll
    for (int v = 0; v < 8; ++v)
      hB[(v + 8 * kh) * 132 + col0 + n] = gelu_f(c[v] + bv);
  }
  __syncthreads();

  if (wave < OO / 16) {  // whole-wave uniform branch; EXEC stays all-ones
    v8f c2 = {};
    c2 = wmma_acc<HH>(hB, 132, W2 + col0, OO, c2);
    const float bv = b2[col0 + n];
    float* op = obuf + (size_t)rc * NN * OO;
#pragma unroll
    for (int v = 0; v < 8; ++v)
      op[(size_t)(row0 + v + 8 * kh) * OO + col0 + n] = c2[v] + bv;
  }
}

// ---------------------------------------------------------------------------
// Kernel 6a/6b: masked pooling (atomics), 6c: divide
// ---------------------------------------------------------------------------
__global__ __launch_bounds__(256) void pool_num_kernel(
    const float* __restrict__ obuf, const float* __restrict__ mask,
    const int* __restrict__ batch, float* __restrict__ num) {
  const long long tid = (long long)blockIdx.x * 256 + threadIdx.x;
  if (tid >= (long long)RC * NN * (OO / 4)) return;
  const int o4 = (int)(tid & 7);
  const int nd = (int)((tid >> 3) % NN);
  const int rc = (int)((tid >> 3) / NN);
  const float mk = mask[(size_t)rc * NN + nd];
  const int g = batch[nd];
  const float4 v =
      *(const float4*)(obuf + ((size_t)rc * NN + nd) * OO + o4 * 4);
  float* np = num + ((size_t)rc * GG + g) * OO + o4 * 4;
  unsafeAtomicAdd(np + 0, v.x * mk);
  unsafeAtomicAdd(np + 1, v.y * mk);
  unsafeAtomicAdd(np + 2, v.z * mk);
  unsafeAtomicAdd(np + 3, v.w * mk);
}

__global__ __launch_bounds__(256) void pool_den_kernel(
    const float* __restrict__ mask, const int* __restrict__ batch,
    float* __restrict__ den) {
  const int tid = blockIdx.x * 256 + threadIdx.x;
  if (tid >= RC * NN) return;
  const int nd = tid % NN;
  const int rc = tid / NN;
  unsafeAtomicAdd(den + (size_t)rc * GG + batch[nd],
                  mask[(size_t)rc * NN + nd]);
}

__global__ __launch_bounds__(256) void finalize_kernel(
    float* __restrict__ out, const float* __restrict__ den) {
  const int i = blockIdx.x * 256 + threadIdx.x;
  if (i >= RC * GG * OO) return;
  const int g = (i / OO) % GG;
  const int rc = i / (OO * GG);
  out[i] = out[i] / (den[(size_t)rc * GG + g] + 1e-7f);
}

// ---------------------------------------------------------------------------
// Host-side orchestration (graph-capturable: only stream ops)
// ---------------------------------------------------------------------------
extern "C" void kernel_launch(void* const* d_in, const int* in_sizes, int n_in,
                              void* d_out, int out_size, void* d_ws,
                              size_t ws_size, hipStream_t stream) {
  const float* x_feat = (const float*)d_in[0];
  const int* eidx = (const int*)d_in[1];
  const float* eattr = (const float*)d_in[2];
  const int* batch = (const int*)d_in[3];
  const float* nmask = (const float*)d_in[4];
  const float* atomW = (const float*)d_in[5];
  const float* atomB = (const float*)d_in[6];
  const float* eps = (const float*)d_in[7];
  const float* bondW = (const float*)d_in[8];
  const float* bondB = (const float*)d_in[9];
  const float* m1W = (const float*)d_in[10];
  const float* m1B = (const float*)d_in[11];
  const float* m2W = (const float*)d_in[12];
  const float* m2B = (const float*)d_in[13];
  const float* h1W = (const float*)d_in[14];
  const float* h1B = (const float*)d_in[15];
  const float* h2W = (const float*)d_in[16];
  const float* h2B = (const float*)d_in[17];
  float* out = (float*)d_out;

  const int* src = eidx;
  const int* dst = eidx + EE;

  float* ux = (float*)d_ws;                      // RC*NN*HH
  float* agg = ux + (size_t)RC * NN * HH;        // RC*NN*HH
  float* eemb = agg + (size_t)RC * NN * HH;      // EE*HH
  float* obuf = eemb + (size_t)EE * HH;          // RC*NN*OO
  float* den = obuf + (size_t)RC * NN * OO;      // RC*GG

  // atom encoder -> ux (broadcast to all RC replicas)
  atom_gemm_kernel<<<NN / 16, 256, 0, stream>>>(x_feat, atomW, atomB, ux);

  for (int l = 0; l < NLAY; ++l) {
    hipMemsetAsync(agg, 0, (size_t)RC * NN * HH * sizeof(float), stream);
    edge_gemm_kernel<<<EE / 16, 256, 0, stream>>>(
        eattr, bondW + (size_t)l * FE * HH, bondB + (size_t)l * HH, eemb);
    msg_scatter_kernel<<<(RC * EE * (HH / 4)) / 256, 256, 0, stream>>>(
        ux, eemb, src, dst, nmask, agg);
    gin_mlp_kernel<<<RC * (NN / 16), 256, 0, stream>>>(
        ux, agg, eps + l, m1W + (size_t)l * HH * HH, m1B + (size_t)l * HH,
        m2W + (size_t)l * HH * HH, m2B + (size_t)l * HH);
  }

  head_mlp_kernel<<<RC * (NN / 16), 256, 0, stream>>>(ux, h1W, h1B, h2W, h2B,
                                                      obuf);

  hipMemsetAsync(out, 0, (size_t)RC * GG * OO * sizeof(float), stream);
  hipMemsetAsync(den, 0, (size_t)RC * GG * sizeof(float), stream);
  pool_num_kernel<<<(RC * NN * (OO / 4)) / 256, 256, 0, stream>>>(obuf, nmask,
                                                                  batch, out);
  pool_den_kernel<<<(RC * NN) / 256, 256, 0, stream>>>(nmask, batch, den);
  finalize_kernel<<<(RC * GG * OO) / 256, 256, 0, stream>>>(out, den);
}